// CNN_invariant_28226525069678
// MI455X (gfx1250) — compile-verified
//
#include <hip/hip_runtime.h>
#include <math.h>

// ---------------- problem constants ----------------
#define IN_C   16
#define OUT_C  16
#define KW     9
#define JK     (IN_C * KW)            // 144
#define NPTS   16384                  // 128*128
#define MPP    (OUT_C * IN_C * KW)    // 2304 mask floats per point

// ---------------- tiling ----------------
#define PT           8                       // points per tile
#define TILES_PER_WG 8
#define PTS_PER_WG   (PT * TILES_PER_WG)     // 64
#define NWG          (NPTS / PTS_PER_WG)     // 256
#define TILE_FLOATS  (PT * MPP)              // 18432
#define BLOCK        128                     // 4 waves (wave32)

typedef float v2f  __attribute__((ext_vector_type(2)));
typedef float v8f  __attribute__((ext_vector_type(8)));
typedef unsigned int u32x4 __attribute__((ext_vector_type(4)));
typedef int   i32x4 __attribute__((ext_vector_type(4)));
typedef int   i32x8 __attribute__((ext_vector_type(8)));

typedef __attribute__((address_space(3))) char lds_char;

// byte offset of a __shared__ object within the workgroup LDS allocation
static __device__ __forceinline__ unsigned lds_off(const void* p) {
  return (unsigned)(unsigned long long)(lds_char*)p;
}

// One TDM descriptor: contiguous 1-D copy of TILE_FLOATS f32 from global -> LDS.
// D# group0: count=1 | lds_addr | global_addr(57b) | type=2
// D# group1: data_size=4B, tensor_dim0=tile_dim0=TILE_FLOATS, 1 row.
static __device__ __forceinline__ void tdm_load_tile(unsigned lds_byte_off,
                                                     const float* gsrc) {
  unsigned long long ga = (unsigned long long)gsrc;
  u32x4 g0;
  g0.x = 1u;                                           // count = 1 (valid D#)
  g0.y = lds_byte_off;                                 // lds_addr
  g0.z = (unsigned)(ga & 0xFFFFFFFFull);               // global_addr[31:0]
  g0.w = (unsigned)((ga >> 32) & 0x1FFFFFFull)         // global_addr[56:32]
       | (2u << 30);                                   // type = 2 ("image")
  i32x8 g1;
  g1[0] = (int)(2u << 16);                                        // data_size=2 (4B)
  g1[1] = (int)(((unsigned)TILE_FLOATS & 0xFFFFu) << 16);          // tensor_dim0[15:0]
  g1[2] = (int)((((unsigned)TILE_FLOATS >> 16) & 0xFFFFu)          // tensor_dim0[31:16]
       | (1u << 16));                                              // tensor_dim1 = 1
  g1[3] = (int)(((unsigned)TILE_FLOATS) << 16);                    // tile_dim0 (fits 16b)
  g1[4] = 0;                                                       // tile_dim1/2 unused
  g1[5] = (int)TILE_FLOATS;                                        // dim0_stride[31:0]
  g1[6] = 0;
  g1[7] = 0;
  i32x4 z4 = {0, 0, 0, 0};
  i32x8 z8 = {0, 0, 0, 0, 0, 0, 0, 0};
  // amdgpu-toolchain (clang-23) 6-arg form: (g0, g1, g2, g3, extra, cpol)
  __builtin_amdgcn_tensor_load_to_lds(g0, g1, z4, z4, z8, 0);
}

__global__ __launch_bounds__(BLOCK)
void cnn_masked_conv_kernel(const float* __restrict__ x,     // (16, N)
                            const float* __restrict__ W,     // (16,16,9)
                            const float* __restrict__ b,     // (16)
                            const float* __restrict__ mask,  // (N,16,16,9)
                            const int*   __restrict__ ker,   // (N,9)
                            float*       __restrict__ out)   // (16, N)
{
  __shared__ float w_s[MPP];            // W, staged once:      9216 B
  __shared__ float xg_s[PT * JK];       // gathered x per tile: 4608 B
  __shared__ float mask_s[2][TILE_FLOATS]; // ping-pong mask:  2x73728 B

  const int tid  = threadIdx.x;
  const int wid  = tid >> 5;            // wave id (wave32)
  const int lane = tid & 31;
  const int half = lane >> 4;           // K-half for WMMA A/B operand layout
  const int li   = lane & 15;           // output channel i handled by this lane
  const int wg_pbase = blockIdx.x * PTS_PER_WG;

  // stage W (shared by every point)
  for (int q = tid; q < MPP; q += BLOCK) w_s[q] = W[q];

  // kick off TDM for tile 0 (one wave issues; TDM is per-wave, EXEC-independent)
  if (wid == 0) {
    tdm_load_tile(lds_off(&mask_s[0][0]),
                  mask + (size_t)wg_pbase * MPP);
  }

  for (int t = 0; t < TILES_PER_WG; ++t) {
    const int buf   = t & 1;
    const int pbase = wg_pbase + t * PT;

    if (wid == 0) {
      if (t + 1 < TILES_PER_WG) {
        // prefetch next tile into the other buffer, then wait for current one
        tdm_load_tile(lds_off(&mask_s[buf ^ 1][0]),
                      mask + (size_t)(pbase + PT) * MPP);
        __builtin_amdgcn_s_wait_tensorcnt(1);   // tile t landed (in-order TDM)
      } else {
        __builtin_amdgcn_s_wait_tensorcnt(0);
      }
    }
    __syncthreads();   // mask tile t visible to all waves; xg_s free for reuse

    // gather xg_s[p][j*9+k] = x[j, ker[pbase+p, k]]  (x is L2-resident)
    {
      const int p = tid >> 4;           // 0..7
      const int j = tid & 15;           // 0..15
      const int*  kr   = ker + (size_t)(pbase + p) * KW;
      const float* xrow = x + (size_t)j * NPTS;
      float* dst = &xg_s[p * JK + j * KW];
      #pragma unroll
      for (int k = 0; k < KW; ++k) dst[k] = xrow[kr[k]];
    }
    __syncthreads();   // xg ready

    // each wave reduces 2 points via 36 chained V_WMMA_F32_16X16X4_F32
    for (int pp = 0; pp < 2; ++pp) {
      const int p = 2 * wid + pp;
      const float* mrow = &mask_s[buf][p * MPP + li * JK];  // mask[p, i=li, :]
      const float* wrow = &w_s[li * JK];                    // W[i=li, :]
      const float* xgp  = &xg_s[p * JK];

      v8f acc = {0.f, 0.f, 0.f, 0.f, 0.f, 0.f, 0.f, 0.f};
      #pragma unroll 4
      for (int tt = 0; tt < JK / 4; ++tt) {
        const int q0 = 4 * tt + 2 * half;   // this lane-half's K pair
        v2f A, B;
        A.x = xgp[q0];                      // A rows replicated: A[m,kk]=xg[q]
        A.y = xgp[q0 + 1];
        B.x = mrow[q0]     * wrow[q0];      // B[kk, n=i] = W*mask (masked weight)
        B.y = mrow[q0 + 1] * wrow[q0 + 1];
        acc = __builtin_amdgcn_wmma_f32_16x16x4_f32(
                  false, A, false, B, (short)0, acc, false, false);
      }
      // every row of D equals y[p,:]; lane n holds column n => y[p, li]
      if (half == 0) {
        float y = acc[0] + b[li];
        float r = (y > 0.f) ? y : (expf(y) - 1.f);   // ELU (alpha=1)
        out[(size_t)li * NPTS + (pbase + p)] = r;
      }
    }
    __syncthreads();   // done reading mask_s[buf] & xg_s before they are reused
  }
}

extern "C" void kernel_launch(void* const* d_in, const int* in_sizes, int n_in,
                              void* d_out, int out_size, void* d_ws, size_t ws_size,
                              hipStream_t stream) {
  const float* x    = (const float*)d_in[0];
  const float* W    = (const float*)d_in[1];
  const float* b    = (const float*)d_in[2];
  const float* mask = (const float*)d_in[3];
  const int*   ker  = (const int*)d_in[4];
  float* out = (float*)d_out;
  (void)in_sizes; (void)n_in; (void)out_size; (void)d_ws; (void)ws_size;

  cnn_masked_conv_kernel<<<NWG, BLOCK, 0, stream>>>(x, W, b, mask, ker, out);
}